// HybridTextEncoder_84722524881478
// MI455X (gfx1250) — compile-verified
//
#include <hip/hip_runtime.h>

typedef float v2f __attribute__((ext_vector_type(2)));
typedef float v8f __attribute__((ext_vector_type(8)));

#define NROWS 32768L   // B*T
#define TLEN  4096
#define NB    8

#if defined(__HIP_DEVICE_COMPILE__) && \
    __has_builtin(__builtin_amdgcn_global_load_async_to_lds_b128) && \
    __has_builtin(__builtin_amdgcn_s_wait_asynccnt)
#define ASYNC_LDS 1
typedef int b128_t __attribute__((vector_size(16)));
typedef __attribute__((address_space(1))) b128_t* gptr128;
typedef __attribute__((address_space(3))) b128_t* lptr128;
#else
#define ASYNC_LDS 0
#endif

static __device__ __forceinline__ v8f wmma4(v2f a, v2f b, v8f c) {
    return __builtin_amdgcn_wmma_f32_16x16x4_f32(false, a, false, b, (short)0, c, false, false);
}

// ---------------------------------------------------------------- embedding gather
__global__ __launch_bounds__(256) void gather_embed(const int* __restrict__ tok,
                                                    const float* __restrict__ emb,
                                                    float* __restrict__ x) {
    int idx = blockIdx.x * 256 + threadIdx.x;     // 32768*64 lanes
    int ti = idx >> 6, l = idx & 63;
    int t = tok[ti];
    const float4* s = (const float4*)(emb + (long)t * 256);
    ((float4*)(x + (long)ti * 256))[l] = s[l];
}

// ---------------------------------------------------------------- row mean (u0)
__global__ __launch_bounds__(256) void row_mean(const float* __restrict__ x,
                                                float* __restrict__ u) {
    int lane = threadIdx.x & 31, wave = threadIdx.x >> 5;
    long row = (long)blockIdx.x * 8 + wave;
    const float* p = x + row * 256;
    float s = 0.f;
#pragma unroll
    for (int i = 0; i < 8; ++i) s += p[lane + i * 32];
#pragma unroll
    for (int off = 16; off; off >>= 1) s += __shfl_xor(s, off, 32);
    if (lane == 0) u[row] = s * (1.f / 256.f);
}

// ---------------------------------------------------------------- generic f32 WMMA GEMM
// C[M,N] = A[M,K] @ B.  BMODE 0: B row-major [K][N].  BMODE 1: B = W[N][K] (use W^T).
// EPI 1: += bias[n] + resid[m*ldc+n]
template <int BMODE, int EPI>
__global__ __launch_bounds__(128) void wmma_gemm_f32(
    const float* __restrict__ A, const float* __restrict__ B, float* __restrict__ C,
    int M, int N, int K, int lda, int ldb, int ldc,
    long abatch, long bbatch, long cbatch,
    const float* __restrict__ bias, const float* __restrict__ resid) {
    int z = blockIdx.z;
    A += (long)z * abatch; B += (long)z * bbatch; C += (long)z * cbatch;
    int lane = threadIdx.x & 31, wave = threadIdx.x >> 5;
    int m0 = (blockIdx.y * 4 + wave) * 16;
    int n0 = blockIdx.x * 16;
    if (m0 >= M || n0 >= N) return;
    int l16 = lane & 15, h = lane >> 4;
    const float* arow = A + (long)(m0 + l16) * lda + 2 * h;
    v8f acc = {0.f, 0.f, 0.f, 0.f, 0.f, 0.f, 0.f, 0.f};
    if (BMODE == 1) {
        const float* brow = B + (long)(n0 + l16) * ldb + 2 * h;
        for (int k = 0; k < K; k += 4) {
            v2f a = *(const v2f*)(arow + k);
            v2f b = *(const v2f*)(brow + k);
            acc = wmma4(a, b, acc);
        }
    } else {
        const float* bcol = B + n0 + l16;
        for (int k = 0; k < K; k += 4) {
            v2f a = *(const v2f*)(arow + k);
            v2f b;
            b[0] = bcol[(long)(k + 2 * h) * ldb];
            b[1] = bcol[(long)(k + 2 * h + 1) * ldb];
            acc = wmma4(a, b, acc);
        }
    }
    int n = n0 + l16;
    float badd = (EPI == 1) ? bias[n] : 0.f;
#pragma unroll
    for (int g = 0; g < 8; ++g) {
        long m = m0 + g + 8 * h;
        float v = acc[g] + badd;
        if (EPI == 1) v += resid[m * ldc + n];
        C[m * ldc + n] = v;
    }
}

// ---------------------------------------------------------------- SSM kernel vectors
// w_i = (A^T)^i c  (i=0..63),  v_m = (A^64)^m b  (m=0..63)
__global__ __launch_bounds__(256) void ssm_vecs(const float* __restrict__ Aall,
                                                const float* __restrict__ Bv,
                                                const float* __restrict__ Cv,
                                                const float* __restrict__ M64,
                                                float* __restrict__ Wv,
                                                float* __restrict__ Vv) {
    int l = blockIdx.x, tid = threadIdx.x;
    const float* A = Aall + (long)l * 65536;
    const float* Mm = M64 + (long)l * 65536;
    __shared__ float vec[256];
    vec[tid] = Cv[l * 256 + tid];
    __syncthreads();
    Wv[((long)l * 64 + 0) * 256 + tid] = vec[tid];
    for (int i = 1; i < 64; ++i) {
        float acc = 0.f;
        for (int r = 0; r < 256; ++r) acc += A[r * 256 + tid] * vec[r];
        __syncthreads();
        vec[tid] = acc;
        __syncthreads();
        Wv[((long)l * 64 + i) * 256 + tid] = acc;
    }
    __syncthreads();
    vec[tid] = Bv[l * 256 + tid];
    __syncthreads();
    Vv[((long)l * 64 + 0) * 256 + tid] = vec[tid];
    for (int m = 1; m < 64; ++m) {
        float acc = 0.f;
        for (int k = 0; k < 256; ++k) acc += Mm[tid * 256 + k] * vec[k];
        __syncthreads();
        vec[tid] = acc;
        __syncthreads();
        Vv[((long)l * 64 + m) * 256 + tid] = acc;
    }
}

// k[l][64m+i] = w_i . v_m
__global__ __launch_bounds__(256) void ssm_kker(const float* __restrict__ Wv,
                                                const float* __restrict__ Vv,
                                                float* __restrict__ kk) {
    int l = blockIdx.y;
    int t = blockIdx.x * 256 + threadIdx.x;
    int i = t & 63, m = t >> 6;
    const float* w = Wv + ((long)l * 64 + i) * 256;
    const float* v = Vv + ((long)l * 64 + m) * 256;
    float acc = 0.f;
    for (int e = 0; e < 256; ++e) acc += w[e] * v[e];
    kk[(long)l * TLEN + t] = acc;
}

// ---------------------------------------------------------------- causal conv: u_next = (1+d)u + k (*) u
__global__ __launch_bounds__(256) void ssm_conv(const float* __restrict__ kl,
                                                const float* __restrict__ ucur,
                                                float* __restrict__ unext,
                                                const float* __restrict__ Dv, int l) {
    __shared__ float ks[512];
    __shared__ float us[1024];
    int tc = blockIdx.x, b = blockIdx.y, tid = threadIdx.x;
    int t0 = tc * 512;
    int t1 = t0 + tid, t2 = t1 + 256;
    const float* ub = ucur + (long)b * TLEN;
    float a1 = 0.f, a2 = 0.f;
    for (int jc = 0; jc <= tc; ++jc) {
        int j0 = jc * 512;
        ks[tid] = kl[j0 + tid];
        ks[tid + 256] = kl[j0 + tid + 256];
        int base = t0 - j0 - 511;
#pragma unroll
        for (int q = 0; q < 4; ++q) {
            int idx = tid + q * 256;
            int gi = base + idx;
            us[idx] = (gi >= 0 && gi < TLEN) ? ub[gi] : 0.f;
        }
        __syncthreads();
        int jm1 = (jc < tc) ? 511 : tid;
        int jm2 = (jc < tc) ? 511 : ((tid + 256 < 511) ? tid + 256 : 511);
        int o1 = tid + 511, o2 = tid + 767;
        for (int jj = 0; jj <= jm1; ++jj) a1 += ks[jj] * us[o1 - jj];
        for (int jj = 0; jj <= jm2; ++jj) a2 += ks[jj] * us[o2 - jj];
        __syncthreads();
    }
    float d1 = 1.f + Dv[l];
    unext[(long)b * TLEN + t1] = d1 * ub[t1] + a1;
    unext[(long)b * TLEN + t2] = d1 * ub[t2] + a2;
}

// ---------------------------------------------------------------- dilated grouped convs -> ms [b][t][c]
__global__ __launch_bounds__(256) void dconv(const float* __restrict__ x,
                                             const float* __restrict__ cw,
                                             const float* __restrict__ cb,
                                             float* __restrict__ ms) {
    __shared__ float xs[48 * 256];
    int tb = blockIdx.x, b = blockIdx.y, tid = threadIdx.x;
    int t0 = tb * 16;
    const float4* xg = (const float4*)(x + (long)b * TLEN * 256);
    float4* xs4 = (float4*)xs;
#if ASYNC_LDS
    // CDNA5 async global->LDS staging (tracked with ASYNCcnt); zero-pad halo rows
    // with plain LDS stores on the complementary lanes (disjoint addresses).
#pragma unroll
    for (int i = 0; i < 12; ++i) {
        int fi = tid + i * 256;             // float4 index; row=fi/64 col=fi%64
        int gr = t0 - 16 + (fi >> 6);
        if (gr >= 0 && gr < TLEN) {
            __builtin_amdgcn_global_load_async_to_lds_b128(
                (gptr128)(xg + (long)gr * 64 + (fi & 63)),
                (lptr128)(xs4 + fi), 0, 0);
        } else {
            float4 zv = {0.f, 0.f, 0.f, 0.f};
            xs4[fi] = zv;
        }
    }
    __builtin_amdgcn_s_wait_asynccnt(0);
#else
#pragma unroll
    for (int i = 0; i < 12; ++i) {
        int fi = tid + i * 256;             // float4 index; row=fi/64 col=fi%64
        int gr = t0 - 16 + (fi >> 6);
        float4 val = {0.f, 0.f, 0.f, 0.f};
        if (gr >= 0 && gr < TLEN) val = xg[(long)gr * 64 + (fi & 63)];
        xs4[fi] = val;
    }
#endif
    __syncthreads();
    int c = tid;
    int idil = c >> 6, oc = c & 63, g = oc >> 4;
    int d = 1 << idil;
    const float* w = cw + (long)(idil * 64 + oc) * 192;   // [4][64][64][3]
    float acc[16];
    float bb = cb[idil * 64 + oc];
#pragma unroll
    for (int t = 0; t < 16; ++t) acc[t] = bb;
    const float* colbase = xs + g * 64;
    for (int ic = 0; ic < 64; ++ic) {
        float w0 = w[ic * 3 + 0], w1 = w[ic * 3 + 1], w2 = w[ic * 3 + 2];
        const float* col = colbase + ic;
#pragma unroll
        for (int t = 0; t < 16; ++t) {
            int r = t + 16;
            acc[t] += w0 * col[(r - d) * 256] + w1 * col[r * 256] + w2 * col[(r + d) * 256];
        }
    }
    float* o = ms + ((long)b * TLEN + t0) * 256 + c;
#pragma unroll
    for (int t = 0; t < 16; ++t) o[(long)t * 256] = acc[t];
}

// ---------------------------------------------------------------- LayerNorm rows of 256
__global__ __launch_bounds__(256) void ln_rows(float* __restrict__ bw,
                                               const float* __restrict__ g,
                                               const float* __restrict__ bta) {
    int lane = threadIdx.x & 31, wave = threadIdx.x >> 5;
    long row = (long)blockIdx.x * 8 + wave;
    float* p = bw + row * 256;
    float v[8], s = 0.f, s2 = 0.f;
#pragma unroll
    for (int i = 0; i < 8; ++i) {
        v[i] = p[lane + i * 32];
        s += v[i];
        s2 += v[i] * v[i];
    }
#pragma unroll
    for (int off = 16; off; off >>= 1) {
        s += __shfl_xor(s, off, 32);
        s2 += __shfl_xor(s2, off, 32);
    }
    float mu = s * (1.f / 256.f);
    float var = s2 * (1.f / 256.f) - mu * mu;
    float inv = rsqrtf(var + 1e-5f);
#pragma unroll
    for (int i = 0; i < 8; ++i) {
        int cc = lane + i * 32;
        p[cc] = (v[i] - mu) * inv * g[cc] + bta[cc];
    }
}

// ---------------------------------------------------------------- fused MLP: silu(comb@W1^T+b1)@W2^T+b2
__global__ __launch_bounds__(256) void fused_mlp(const float* __restrict__ x,
                                                 const float* __restrict__ bw,
                                                 const float* __restrict__ u0,
                                                 const float* __restrict__ u3,
                                                 const float* __restrict__ W1,
                                                 const float* __restrict__ b1,
                                                 const float* __restrict__ W2,
                                                 const float* __restrict__ b2,
                                                 float* __restrict__ out) {
    __shared__ float h1s[16 * 516];       // 16 rows, +4 pad to avoid bank conflicts
    int lane = threadIdx.x & 31, wave = threadIdx.x >> 5;
    int l16 = lane & 15, h = lane >> 4;
    int m0 = blockIdx.x * 16;
    long row = m0 + l16;
    float ys = u3[row] - u0[row];
    const float* xr = x + row * 256 + 2 * h;
    const float* br = bw + row * 256 + 2 * h;
    const float* w1p[4];
#pragma unroll
    for (int s = 0; s < 4; ++s)
        w1p[s] = W1 + (long)((wave * 4 + s) * 16 + l16) * 512 + 2 * h;
    v8f acc[4];
#pragma unroll
    for (int s = 0; s < 4; ++s) acc[s] = (v8f){0.f, 0.f, 0.f, 0.f, 0.f, 0.f, 0.f, 0.f};
    for (int k = 0; k < 256; k += 4) {          // first half: ff = x + ysum
        v2f a = *(const v2f*)(xr + k);
        a[0] += ys; a[1] += ys;
#pragma unroll
        for (int s = 0; s < 4; ++s) {
            v2f b = *(const v2f*)(w1p[s] + k);
            acc[s] = wmma4(a, b, acc[s]);
        }
    }
    for (int k = 0; k < 256; k += 4) {          // second half: bw
        v2f a = *(const v2f*)(br + k);
#pragma unroll
        for (int s = 0; s < 4; ++s) {
            v2f b = *(const v2f*)(w1p[s] + 256 + k);
            acc[s] = wmma4(a, b, acc[s]);
        }
    }
#pragma unroll
    for (int s = 0; s < 4; ++s) {
        int n = (wave * 4 + s) * 16 + l16;
        float bb = b1[n];
#pragma unroll
        for (int g = 0; g < 8; ++g) {
            float v = acc[s][g] + bb;
            v = v / (1.f + __expf(-v));          // silu
            h1s[(g + 8 * h) * 516 + n] = v;
        }
    }
    __syncthreads();
    v8f o0 = {0.f, 0.f, 0.f, 0.f, 0.f, 0.f, 0.f, 0.f};
    v8f o1 = o0;
    const float* a2 = h1s + l16 * 516 + 2 * h;
    int n0a = (wave * 2 + 0) * 16, n0b = (wave * 2 + 1) * 16;
    const float* w2a = W2 + (long)(n0a + l16) * 512 + 2 * h;
    const float* w2b = W2 + (long)(n0b + l16) * 512 + 2 * h;
    for (int k = 0; k < 512; k += 4) {
        v2f a = *(const v2f*)(a2 + k);
        v2f ba = *(const v2f*)(w2a + k);
        v2f bb = *(const v2f*)(w2b + k);
        o0 = wmma4(a, ba, o0);
        o1 = wmma4(a, bb, o1);
    }
    {
        int n = n0a + l16;
        float bb = b2[n];
#pragma unroll
        for (int g = 0; g < 8; ++g) out[(long)(m0 + g + 8 * h) * 256 + n] = o0[g] + bb;
    }
    {
        int n = n0b + l16;
        float bb = b2[n];
#pragma unroll
        for (int g = 0; g < 8; ++g) out[(long)(m0 + g + 8 * h) * 256 + n] = o1[g] + bb;
    }
}

// ---------------------------------------------------------------- host launcher
extern "C" void kernel_launch(void* const* d_in, const int* in_sizes, int n_in,
                              void* d_out, int out_size, void* d_ws, size_t ws_size,
                              hipStream_t stream) {
    (void)in_sizes; (void)n_in; (void)out_size; (void)ws_size;
    const int* tok = (const int*)d_in[0];
    const float* emb = (const float*)d_in[1];
    const float* Amat = (const float*)d_in[2];
    const float* Bv = (const float*)d_in[3];
    const float* Cv = (const float*)d_in[4];
    const float* Dv = (const float*)d_in[5];
    const float* cw = (const float*)d_in[6];
    const float* cb = (const float*)d_in[7];
    const float* pww = (const float*)d_in[8];
    const float* pwb = (const float*)d_in[9];
    const float* lng = (const float*)d_in[10];
    const float* lnb = (const float*)d_in[11];
    const float* W1 = (const float*)d_in[12];
    const float* b1 = (const float*)d_in[13];
    const float* W2 = (const float*)d_in[14];
    const float* b2 = (const float*)d_in[15];
    float* out = (float*)d_out;

    char* ws = (char*)d_ws;
    size_t off = 0;
    auto take = [&](size_t bytes) -> float* {
        float* p = (float*)(ws + off);
        off = (off + bytes + 255) & ~(size_t)255;
        return p;
    };
    float* xbuf = take(NROWS * 256 * 4);   // [b,t,e]
    float* msbuf = take(NROWS * 256 * 4);  // [b,t,c]
    float* bwbuf = take(NROWS * 256 * 4);
    float* U = take(4 * NROWS * 4);        // u0..u3
    float* Pa = take(3L * 65536 * 4);
    float* Pb = take(3L * 65536 * 4);
    float* Wv = take(3L * 64 * 256 * 4);
    float* Vv = take(3L * 64 * 256 * 4);
    float* kk = take(3L * TLEN * 4);

    gather_embed<<<8192, 256, 0, stream>>>(tok, emb, xbuf);
    row_mean<<<4096, 256, 0, stream>>>(xbuf, U);

    // A^2 .. A^64 via 6 batched WMMA squarings (z = layer)
    dim3 gsq(16, 4, 3);
    wmma_gemm_f32<0, 0><<<gsq, 128, 0, stream>>>(Amat, Amat, Pa, 256, 256, 256, 256, 256, 256,
                                                 65536, 65536, 65536, nullptr, nullptr);
    wmma_gemm_f32<0, 0><<<gsq, 128, 0, stream>>>(Pa, Pa, Pb, 256, 256, 256, 256, 256, 256,
                                                 65536, 65536, 65536, nullptr, nullptr);
    wmma_gemm_f32<0, 0><<<gsq, 128, 0, stream>>>(Pb, Pb, Pa, 256, 256, 256, 256, 256, 256,
                                                 65536, 65536, 65536, nullptr, nullptr);
    wmma_gemm_f32<0, 0><<<gsq, 128, 0, stream>>>(Pa, Pa, Pb, 256, 256, 256, 256, 256, 256,
                                                 65536, 65536, 65536, nullptr, nullptr);
    wmma_gemm_f32<0, 0><<<gsq, 128, 0, stream>>>(Pb, Pb, Pa, 256, 256, 256, 256, 256, 256,
                                                 65536, 65536, 65536, nullptr, nullptr);
    wmma_gemm_f32<0, 0><<<gsq, 128, 0, stream>>>(Pa, Pa, Pb, 256, 256, 256, 256, 256, 256,
                                                 65536, 65536, 65536, nullptr, nullptr);
    // Pb = A^64 per layer

    ssm_vecs<<<3, 256, 0, stream>>>(Amat, Bv, Cv, Pb, Wv, Vv);
    ssm_kker<<<dim3(16, 3), 256, 0, stream>>>(Wv, Vv, kk);
    for (int l = 0; l < 3; ++l)
        ssm_conv<<<dim3(8, 8), 256, 0, stream>>>(kk + (long)l * TLEN, U + (long)l * NROWS,
                                                 U + (long)(l + 1) * NROWS, Dv, l);

    dconv<<<dim3(256, 8), 256, 0, stream>>>(xbuf, cw, cb, msbuf);

    // bw_pre = ms @ pw_w^T + pw_b + x   (residual), then LayerNorm
    wmma_gemm_f32<1, 1><<<dim3(16, 512, 1), 128, 0, stream>>>(msbuf, pww, bwbuf,
                                                              32768, 256, 256, 256, 256, 256,
                                                              0, 0, 0, pwb, xbuf);
    ln_rows<<<4096, 256, 0, stream>>>(bwbuf, lng, lnb);

    fused_mlp<<<2048, 256, 0, stream>>>(xbuf, bwbuf, U, U + 3L * NROWS, W1, b1, W2, b2, out);
}